// MaxBlurPooling2D_67929202753931
// MI455X (gfx1250) — compile-verified
//
#include <hip/hip_runtime.h>
#include <hip/hip_bf16.h>

// MaxBlurPool2D fused: 2x2 s1 maxpool (-inf pad bottom/right) -> 3x3 binomial
// blur (SAME, zero pad) -> 2x2 s2 average. Blur+avg collapse to a separable
// 4x4 stencil over the max-pool field with taps [1,3,3,1] and scale 1/64.
//
// x:   [16,224,224,128] f32 NHWC  -> out: [16,112,112,128] f32
// Memory-bound (>=514 MB @ 23.3 TB/s ~ 22us floor). Stage 19x19x(32ch) input
// tiles into LDS with CDNA5 async global->LDS copies, compute from LDS,
// stream the output with non-temporal stores.

#define NB   16
#define HH   224
#define WW   224
#define OHS  112
#define OWS  112
#define C4T  32      // 128 channels / 4 per float4
#define XC   19      // x-tile extent (8*2+3)
#define CC4  8       // channel-chunk in float4 units (32 channels)
#define NLD  (XC*XC*CC4)   // 2888 float4 staged per block

typedef float v4f __attribute__((ext_vector_type(4)));

__device__ __forceinline__ v4f v4max(v4f a, v4f b) {
    return __builtin_elementwise_max(a, b);
}

__global__ __launch_bounds__(256)
void maxblurpool_kernel(const v4f* __restrict__ x, v4f* __restrict__ out) {
    __shared__ v4f tile[NLD];   // 46,208 B

    const int tid = threadIdx.x;
    const int bz  = blockIdx.z;
    const int n   = bz >> 2;
    const int c4b = (bz & 3) * CC4;       // channel-chunk base (float4 units)
    const int oh0 = blockIdx.y * 8;
    const int ow0 = blockIdx.x * 8;
    const int gr0 = 2 * oh0 - 1;          // global x row of tile row 0
    const int gc0 = 2 * ow0 - 1;          // global x col of tile col 0

    const float ninf = -__builtin_inff();
    const v4f ninfv = {ninf, ninf, ninf, ninf};
    const v4f zero  = {0.f, 0.f, 0.f, 0.f};

    // ---- stage x tile into LDS (CDNA5 async global->LDS, ASYNCcnt) ----
    const unsigned lds0 = (unsigned)(size_t)(&tile[0]);  // low 32b of flat = LDS offset
    for (int i = tid; i < NLD; i += 256) {
        const int c4l  = i & (CC4 - 1);
        const int rest = i >> 3;
        const int lc   = rest % XC;
        const int lr   = rest / XC;
        const int gr = gr0 + lr;
        const int gc = gc0 + lc;
        const unsigned ldsoff = lds0 + (unsigned)i * 16u;
        if ((unsigned)gr < (unsigned)HH && (unsigned)gc < (unsigned)WW) {
            const v4f* gsrc =
                x + ((((size_t)n * HH + gr) * WW + gc) * C4T + (c4b + c4l));
            asm volatile("global_load_async_to_lds_b128 %0, %1, off"
                         :: "v"(ldsoff),
                            "v"((unsigned long long)(size_t)gsrc)
                         : "memory");
        } else {
            tile[i] = ninfv;   // -inf pad for the max-pool stage
        }
    }
    asm volatile("s_wait_asynccnt 0" ::: "memory");
    __syncthreads();

    // ---- compute: each thread -> 2 vertically adjacent outputs, 4 channels ----
    const int c4l = tid & 7;                 // float4 channel within chunk
    const int owi = (tid >> 3) & 7;          // output col within tile
    const int ohg = tid >> 6;                // 0..3 -> output-row pair
    const int oh_t = oh0 + 2 * ohg;          // first output row of this thread
    const int lrb  = 4 * ohg;                // x-tile local row base
    const int lcb  = 2 * owi;                // x-tile local col base

    // column masks: mp col valid in [0, W)
    const bool cv0 = (unsigned)(gc0 + lcb + 0) < (unsigned)WW;
    const bool cv1 = (unsigned)(gc0 + lcb + 1) < (unsigned)WW;
    const bool cv2 = (unsigned)(gc0 + lcb + 2) < (unsigned)WW;
    const bool cv3 = (unsigned)(gc0 + lcb + 3) < (unsigned)WW;

    v4f acc0 = zero, acc1 = zero;
    v4f xp0 = tile[(lrb * XC + lcb + 0) * CC4 + c4l];
    v4f xp1 = tile[(lrb * XC + lcb + 1) * CC4 + c4l];
    v4f xp2 = tile[(lrb * XC + lcb + 2) * CC4 + c4l];
    v4f xp3 = tile[(lrb * XC + lcb + 3) * CC4 + c4l];
    v4f xp4 = tile[(lrb * XC + lcb + 4) * CC4 + c4l];

#pragma unroll
    for (int j = 0; j < 6; ++j) {
        const int lrn = lrb + j + 1;
        v4f xc0 = tile[(lrn * XC + lcb + 0) * CC4 + c4l];
        v4f xc1 = tile[(lrn * XC + lcb + 1) * CC4 + c4l];
        v4f xc2 = tile[(lrn * XC + lcb + 2) * CC4 + c4l];
        v4f xc3 = tile[(lrn * XC + lcb + 3) * CC4 + c4l];
        v4f xc4 = tile[(lrn * XC + lcb + 4) * CC4 + c4l];

        // vertical 2-max then horizontal 2-max = 2x2 max-pool values
        v4f vm0 = v4max(xp0, xc0);
        v4f vm1 = v4max(xp1, xc1);
        v4f vm2 = v4max(xp2, xc2);
        v4f vm3 = v4max(xp3, xc3);
        v4f vm4 = v4max(xp4, xc4);

        v4f m0 = cv0 ? v4max(vm0, vm1) : zero;
        v4f m1 = cv1 ? v4max(vm1, vm2) : zero;
        v4f m2 = cv2 ? v4max(vm2, vm3) : zero;
        v4f m3 = cv3 ? v4max(vm3, vm4) : zero;

        // horizontal [1,3,3,1] tap (fp contract -> fma)
        v4f s = (m0 + m3) + 3.0f * (m1 + m2);

        // mp row validity (blur zero-pad): global mp row = gr0 + lrb + j
        const bool rv = (unsigned)(gr0 + lrb + j) < (unsigned)HH;
        s = rv ? s : zero;

        // vertical taps: out0 uses j=0..3, out1 uses j=2..5 (compile-time pruned)
        constexpr float U[4] = {1.f, 3.f, 3.f, 1.f};
        if (j <= 3) acc0 += U[j] * s;
        if (j >= 2) acc1 += U[j - 2] * s;

        xp0 = xc0; xp1 = xc1; xp2 = xc2; xp3 = xc3; xp4 = xc4;
    }

    const float sc = 1.0f / 64.0f;
    acc0 *= sc;
    acc1 *= sc;

    const size_t ob =
        (((size_t)n * OHS + oh_t) * OWS + (ow0 + owi)) * C4T + (c4b + c4l);
    // output is write-once: stream past L2 with non-temporal stores
    __builtin_nontemporal_store(acc0, &out[ob]);
    __builtin_nontemporal_store(acc1, &out[ob + (size_t)OWS * C4T]);
}

extern "C" void kernel_launch(void* const* d_in, const int* in_sizes, int n_in,
                              void* d_out, int out_size, void* d_ws, size_t ws_size,
                              hipStream_t stream) {
    (void)in_sizes; (void)n_in; (void)d_ws; (void)ws_size; (void)out_size;
    const v4f* x = (const v4f*)d_in[0];   // blur/avg kernels are fixed constants
    v4f* out = (v4f*)d_out;
    dim3 grid(OWS / 8, OHS / 8, NB * 4);  // 14 x 14 x 64 blocks
    maxblurpool_kernel<<<grid, 256, 0, stream>>>(x, out);
}